// AttnHead_80496277061712
// MI455X (gfx1250) — compile-verified
//
#include <hip/hip_runtime.h>
#include <math.h>

typedef __attribute__((ext_vector_type(2))) float v2f;
typedef __attribute__((ext_vector_type(8))) float v8f;

constexpr int B = 2, N = 4096, F_IN = 256, F_OUT = 64;
constexpr int ROWS = B * N;           // 8192
constexpr float EPS = 1e-5f, SLOPE = 0.01f;

// ---------------- kernel 0: zero the channel-sum accumulator ----------------
__global__ void zero_sums_kernel(float* __restrict__ chans) {
    if (threadIdx.x < 2 * F_OUT) chans[threadIdx.x] = 0.0f;
}

// ---------------- kernel 1: seq_fts = seq @ W1^T  (fp32 WMMA) ----------------
// grid = ROWS/16 blocks, 128 threads (4 waves). Wave w computes cols [16w,16w+16).
__global__ __launch_bounds__(128) void proj_kernel(const float* __restrict__ seq,
                                                   const float* __restrict__ W1,
                                                   float* __restrict__ seq_fts)
{
    __shared__ float sA[16 * 260];                 // 16x256 tile, pitch 260 (bank-conflict free)
    const int row0 = blockIdx.x * 16;

    for (int t = threadIdx.x; t < 16 * 64; t += 128) {
        const int r = t >> 6, c4 = t & 63;
        const float4 v = ((const float4*)(seq + (size_t)(row0 + r) * F_IN))[c4];
        *(float4*)&sA[r * 260 + c4 * 4] = v;
    }
    __syncthreads();

    const int wave = threadIdx.x >> 5, lane = threadIdx.x & 31;
    const int half = lane >> 4, l16 = lane & 15;
    const int n = wave * 16 + l16;                 // output column (= W1 row)

    v8f c = {};
    for (int k = 0; k < F_IN; k += 4) {
        const int ka = k + half * 2;               // A/B fragment: lanes 0-15 -> K=0,1 ; 16-31 -> K=2,3
        v2f a, b;
        a.x = sA[l16 * 260 + ka];
        a.y = sA[l16 * 260 + ka + 1];
        b.x = W1[n * F_IN + ka];                   // B[k][n] = W1[n][k]
        b.y = W1[n * F_IN + ka + 1];
        c = __builtin_amdgcn_wmma_f32_16x16x4_f32(false, a, false, b, (short)0, c, false, false);
    }
    for (int r = 0; r < 8; ++r)                    // C: VGPR r -> row r + half*8, col = n
        seq_fts[(size_t)(row0 + r + half * 8) * F_OUT + n] = c[r];
}

// ---------------- kernel 2: f1/f2 per-node dots ----------------
__global__ void f12_kernel(const float* __restrict__ seq_fts,
                           const float* __restrict__ w2, const float* __restrict__ b2,
                           const float* __restrict__ w3, const float* __restrict__ b3,
                           float* __restrict__ f1, float* __restrict__ f2)
{
    const int row = blockIdx.x * blockDim.x + threadIdx.x;
    if (row >= ROWS) return;
    const float* p = seq_fts + (size_t)row * F_OUT;
    float s1 = 0.0f, s2 = 0.0f;
    for (int nn = 0; nn < F_OUT; ++nn) {
        const float v = p[nn];
        s1 = fmaf(v, w2[nn], s1);
        s2 = fmaf(v, w3[nn], s2);
    }
    f1[row] = s1 + b2[0];
    f2[row] = s2 + b3[0];
}

// ---------------- kernel 3: exact softmax row stats (m_i, s_i) ----------------
// one wave per attention row; float4-vectorized streaming read of bias_mat (pass 1 of 2)
__global__ __launch_bounds__(256) void stats_kernel(const float* __restrict__ f1,
                                                    const float* __restrict__ f2,
                                                    const float* __restrict__ bias,
                                                    float* __restrict__ mrow,
                                                    float* __restrict__ srow)
{
    const int gw = (blockIdx.x * blockDim.x + threadIdx.x) >> 5;   // global row 0..ROWS-1
    const int lane = threadIdx.x & 31;
    const int b = gw / N;
    const float f1i = f1[gw];
    const float* f2b = f2 + b * N;
    const float* biasrow = bias + (size_t)gw * N;

    float m = -__builtin_inff(), s = 0.0f;
    for (int j = lane * 4; j < N; j += 128) {
        const float4 bv = *(const float4*)&biasrow[j];
        const float4 fv = *(const float4*)&f2b[j];
        float xs[4] = { f1i + fv.x, f1i + fv.y, f1i + fv.z, f1i + fv.w };
        float bs[4] = { bv.x, bv.y, bv.z, bv.w };
        #pragma unroll
        for (int u = 0; u < 4; ++u) {
            const float x = xs[u];
            const float l = (x > 0.0f ? x : SLOPE * x) + bs[u];
            const float mn = fmaxf(m, l);
            s = s * __expf(m - mn) + __expf(l - mn);
            m = mn;
        }
    }
    for (int off = 16; off > 0; off >>= 1) {       // log-sum-exp tree combine
        const float mo = __shfl_xor(m, off, 32);
        const float so = __shfl_xor(s, off, 32);
        const float mn = fmaxf(m, mo);
        s = s * __expf(m - mn) + so * __expf(mo - mn);
        m = mn;
    }
    if (lane == 0) { mrow[gw] = m; srow[gw] = s; }
}

// -------- async global->LDS staging of one bias[16 x 64] chunk (8 b128 per lane) --------
__device__ __forceinline__ void stage_bias_async(const float* __restrict__ biasb,
                                                 int jc, float* sbuf, int lane)
{
    #pragma unroll
    for (int it = 0; it < 8; ++it) {
        const int t = lane + it * 32;
        const int r = t >> 4, c4 = t & 15;
        const float* g = &biasb[(size_t)r * N + jc + c4 * 4];
        const unsigned l = (unsigned)(uintptr_t)&sbuf[r * 68 + c4 * 4];  // low 32b = LDS offset
        asm volatile("global_load_async_to_lds_b128 %0, %1, off"
                     :: "v"(l), "v"(g) : "memory");
    }
}

// ---------------- kernel 4: vals = softmax(logits) @ seq_fts  (fp32 WMMA) ----------------
// grid = ROWS/16 blocks, 128 threads (4 waves). Each wave owns a 1024-wide j-range and all
// 64 output cols (4 C tiles). Exact m_i from pass 1 -> no accumulator rescale. bias tiles
// stream through the async global->LDS engine, double buffered against WMMA+exp compute.
__global__ __launch_bounds__(128) void attn_kernel(const float* __restrict__ f1,
                                                   const float* __restrict__ f2,
                                                   const float* __restrict__ bias,
                                                   const float* __restrict__ mrow,
                                                   const float* __restrict__ srow,
                                                   const float* __restrict__ seq_fts,
                                                   float* __restrict__ vals,
                                                   float* __restrict__ chans)
{
    __shared__ float smem[4 * 2 * 16 * 68];        // 2 staging bufs per wave (pitch 68); reused for reduction
    const int tile = blockIdx.x;
    const int b = tile >> 8;                        // 256 tiles per batch
    const int i0 = (tile & 255) * 16;
    const int wave = threadIdx.x >> 5, lane = threadIdx.x & 31;
    const int half = lane >> 4, l16 = lane & 15;
    const int rowg = b * N + i0;

    const float f1i = f1[rowg + l16];               // A-fragment row = lane&15 (both halves)
    const float mi  = mrow[rowg + l16];
    const float* f2b  = f2 + b * N;
    const float* seqf = seq_fts + (size_t)b * N * F_OUT;
    const float* biasb = bias + (size_t)rowg * N;
    float* sb0 = smem + wave * (2 * 16 * 68);
    float* sb1 = sb0 + 16 * 68;

    v8f c0 = {}, c1 = {}, c2 = {}, c3 = {};
    const int jbeg = wave * (N / 4);
    const int jend = jbeg + N / 4;

    stage_bias_async(biasb, jbeg, sb0, lane);       // prologue: chunk 0 -> buf 0
    int buf = 0;
    for (int jc = jbeg; jc < jend; jc += 64) {
        float* scur = buf ? sb1 : sb0;
        if (jc + 64 < jend) {                       // stage next chunk into the other buffer
            stage_bias_async(biasb, jc + 64, buf ? sb0 : sb1, lane);
            asm volatile("s_wait_asynccnt 0x8" ::: "memory");  // current chunk landed (in-order)
        } else {
            asm volatile("s_wait_asynccnt 0x0" ::: "memory");  // drain
        }

        for (int j4 = 0; j4 < 64; j4 += 4) {
            const int jj = jc + j4 + half * 2;      // this lane's K slots
            const int lo = j4 + half * 2;
            // P[i,j] = exp(leaky_relu(f1_i+f2_j)+bias - m_i), in A-fragment layout
            const float x0 = f1i + f2b[jj];
            const float l0 = (x0 > 0.0f ? x0 : SLOPE * x0) + scur[l16 * 68 + lo];
            const float x1 = f1i + f2b[jj + 1];
            const float l1 = (x1 > 0.0f ? x1 : SLOPE * x1) + scur[l16 * 68 + lo + 1];
            v2f a; a.x = __expf(l0 - mi); a.y = __expf(l1 - mi);

            v2f bb;
            bb.x = seqf[(size_t)jj * F_OUT + l16];
            bb.y = seqf[(size_t)(jj + 1) * F_OUT + l16];
            c0 = __builtin_amdgcn_wmma_f32_16x16x4_f32(false, a, false, bb, (short)0, c0, false, false);
            bb.x = seqf[(size_t)jj * F_OUT + 16 + l16];
            bb.y = seqf[(size_t)(jj + 1) * F_OUT + 16 + l16];
            c1 = __builtin_amdgcn_wmma_f32_16x16x4_f32(false, a, false, bb, (short)0, c1, false, false);
            bb.x = seqf[(size_t)jj * F_OUT + 32 + l16];
            bb.y = seqf[(size_t)(jj + 1) * F_OUT + 32 + l16];
            c2 = __builtin_amdgcn_wmma_f32_16x16x4_f32(false, a, false, bb, (short)0, c2, false, false);
            bb.x = seqf[(size_t)jj * F_OUT + 48 + l16];
            bb.y = seqf[(size_t)(jj + 1) * F_OUT + 48 + l16];
            c3 = __builtin_amdgcn_wmma_f32_16x16x4_f32(false, a, false, bb, (short)0, c3, false, false);
        }
        buf ^= 1;
    }

    // cross-wave reduction of the 16x64 partial tiles
    __syncthreads();
    for (int r = 0; r < 8; ++r) {
        const int row = r + half * 8;
        smem[wave * 1024 + row * 64 +      l16] = c0[r];
        smem[wave * 1024 + row * 64 + 16 + l16] = c1[r];
        smem[wave * 1024 + row * 64 + 32 + l16] = c2[r];
        smem[wave * 1024 + row * 64 + 48 + l16] = c3[r];
    }
    __syncthreads();
    for (int t = threadIdx.x; t < 16 * 64; t += 128) {
        const int row = t >> 6, nn = t & 63;
        const float v = smem[t] + smem[1024 + t] + smem[2048 + t] + smem[3072 + t];
        const float sv = v / srow[rowg + row];
        vals[(size_t)(rowg + row) * F_OUT + nn] = sv;
        smem[t] = sv;                               // keep for channel-sum reduce
    }
    __syncthreads();
    if (threadIdx.x < F_OUT) {                      // per-block channel sums -> 2 atomics/channel
        const int nn = threadIdx.x;
        float s = 0.0f, sq = 0.0f;
        for (int row = 0; row < 16; ++row) {
            const float v = smem[row * 64 + nn];
            s += v; sq += v * v;
        }
        atomicAdd(&chans[nn], s);
        atomicAdd(&chans[F_OUT + nn], sq);
    }
}

// ---------------- kernel 5: batchnorm (batch stats, biased var) + ELU ----------------
__global__ void bn_elu_kernel(const float* __restrict__ vals, const float* __restrict__ chans,
                              const float* __restrict__ gamma, const float* __restrict__ beta,
                              float* __restrict__ out)
{
    const int idx = blockIdx.x * blockDim.x + threadIdx.x;
    if (idx >= ROWS * F_OUT) return;
    const int nn = idx & (F_OUT - 1);
    const float inv = 1.0f / (float)ROWS;
    const float mean = chans[nn] * inv;
    const float var  = chans[F_OUT + nn] * inv - mean * mean;
    const float x = (vals[idx] - mean) * rsqrtf(var + EPS) * gamma[nn] + beta[nn];
    out[idx] = x > 0.0f ? x : expm1f(x);
}

extern "C" void kernel_launch(void* const* d_in, const int* in_sizes, int n_in,
                              void* d_out, int out_size, void* d_ws, size_t ws_size,
                              hipStream_t stream)
{
    (void)in_sizes; (void)n_in; (void)out_size; (void)ws_size;
    const float* seq   = (const float*)d_in[0];
    const float* bias  = (const float*)d_in[1];
    const float* W1    = (const float*)d_in[2];
    const float* w2    = (const float*)d_in[3];
    const float* b2    = (const float*)d_in[4];
    const float* w3    = (const float*)d_in[5];
    const float* b3    = (const float*)d_in[6];
    const float* gamma = (const float*)d_in[7];
    const float* beta  = (const float*)d_in[8];
    float* out = (float*)d_out;

    float* ws      = (float*)d_ws;                 // ~4.3 MB total
    float* seq_fts = ws;                           // ROWS*F_OUT
    float* f1      = seq_fts + (size_t)ROWS * F_OUT;
    float* f2      = f1 + ROWS;
    float* mrow    = f2 + ROWS;
    float* srow    = mrow + ROWS;
    float* vals    = srow + ROWS;                  // ROWS*F_OUT
    float* chans   = vals + (size_t)ROWS * F_OUT;  // 2*F_OUT

    zero_sums_kernel<<<1, 128, 0, stream>>>(chans);
    proj_kernel<<<ROWS / 16, 128, 0, stream>>>(seq, W1, seq_fts);
    f12_kernel<<<ROWS / 256, 256, 0, stream>>>(seq_fts, w2, b2, w3, b3, f1, f2);
    stats_kernel<<<(ROWS * 32) / 256, 256, 0, stream>>>(f1, f2, bias, mrow, srow);
    attn_kernel<<<ROWS / 16, 128, 0, stream>>>(f1, f2, bias, mrow, srow, seq_fts, vals, chans);
    bn_elu_kernel<<<(ROWS * F_OUT) / 256, 256, 0, stream>>>(vals, chans, gamma, beta, out);
}